// DualHeadLoss_27479200760241
// MI455X (gfx1250) — compile-verified
//
#include <hip/hip_runtime.h>

// DualHeadLoss: memory-bound streaming reduction.
//   bytes/elem = 4*4 (f32) + 4 (i32) + 1 (bool) = 21 B -> 352 MB total @ 23.3 TB/s ~= 15 us floor.
// Strategy: vec4 non-temporal loads (TH_LOAD_NT, stream > 192MB L2, single use),
// per-thread f32 accumulation, wave reduce via native-f32 V_WMMA_F32_16X16X4_F32
// (ones-B row-sum trick, bit-exact), deterministic two-pass block-partial reduce.

typedef __attribute__((ext_vector_type(2))) float v2f;
typedef __attribute__((ext_vector_type(4))) float v4f;
typedef __attribute__((ext_vector_type(4))) int   v4i;
typedef __attribute__((ext_vector_type(8))) float v8f;

#define NBLOCKS  2048
#define NTHREADS 256
#define NWAVES   (NTHREADS / 32)
#define DELTA_F  1.0f

// ---- wave reduction via f32 WMMA -------------------------------------------
// Pack lane partials as A (16x4 f32): lane L<16 supplies A[L,0] (vgpr0) and
// A[L,1]=0 (vgpr1); lane L>=16 supplies A[L-16,2] and A[L-16,3]=0.
// B = all-ones 4x16 -> D[m,n] = partial[m] + partial[m+16] for every n.
// D layout: lanes 0-15 hold rows 0-7 in c[0..7]; lanes 16-31 hold rows 8-15.
// So after summing c[0..7]:
//   lanes 0-15  : s = sum of partials from lanes {0..7, 16..23}
//   lanes 16-31 : s = sum of partials from lanes {8..15, 24..31}
// Each wave emits TWO half-sums (lane 0 and lane 16) straight to LDS — no
// cross-lane shuffle needed at all.
// EXEC must be all-1s at the WMMA (callers guarantee full reconvergence).
__device__ __forceinline__ float wave_halfsum_wmma(float v) {
  v2f a; a[0] = v;    a[1] = 0.0f;
  v2f b; b[0] = 1.0f; b[1] = 1.0f;
  v8f c = {};
  c = __builtin_amdgcn_wmma_f32_16x16x4_f32(
      /*neg_a=*/false, a, /*neg_b=*/false, b,
      /*c_mod=*/(short)0, c, /*reuse_a=*/false, /*reuse_b=*/false);
  return ((c[0] + c[1]) + (c[2] + c[3])) + ((c[4] + c[5]) + (c[6] + c[7]));
}

// Block reduce; result valid on threadIdx.x == 0. lds must hold 2*NWAVES floats.
__device__ __forceinline__ float block_reduce(float v, float* lds) {
  const float h = wave_halfsum_wmma(v);
  // lane 0 and lane 16 of each wave carry complementary half-wave sums
  if ((threadIdx.x & 15) == 0) {
    lds[threadIdx.x >> 4] = h;   // slot = wave*2 + (lane>=16)
  }
  __syncthreads();
  float s = 0.0f;
  if (threadIdx.x == 0) {
#pragma unroll
    for (int i = 0; i < 2 * NWAVES; ++i) s += lds[i];   // 2x ds_load_b128 x2
  }
  return s;
}

// ---- per-element loss -------------------------------------------------------
__device__ __forceinline__ float loss1(float pe, float ps, float tg,
                                       int id, unsigned cen, float thr) {
  const float pred  = (id == 0) ? pe : ps;
  // censored branch: relu(threshold - pred)^2  (CENSORING_WEIGHT = 1)
  const float cpos  = fmaxf(thr - pred, 0.0f);
  const float closs = cpos * cpos;
  // huber(delta = 1)
  const float ae  = fabsf(pred - tg);
  const float q   = fminf(ae, DELTA_F);
  const float hub = 0.5f * q * q + DELTA_F * (ae - q);
  return cen ? closs : hub;
}

// ---- pass 1: grid-stride stream -> per-block partial sums -------------------
__global__ __launch_bounds__(NTHREADS) void dual_head_partial(
    const float* __restrict__ pe, const float* __restrict__ ps,
    const float* __restrict__ tg, const int* __restrict__ ids,
    const unsigned char* __restrict__ cen, const float* __restrict__ thr,
    float* __restrict__ partials, int n) {
  __shared__ float lds[2 * NWAVES];
  const int tid    = blockIdx.x * NTHREADS + threadIdx.x;
  const int stride = NBLOCKS * NTHREADS;
  const int nv4    = n >> 2;       // N = 2^24 -> 2^22 vec4, 8 iters/thread, no tail

  float acc = 0.0f;
  for (int i = tid; i < nv4; i += stride) {
    // Non-temporal 128-bit loads: single-use stream, keep out of L2.
    v4f a  = __builtin_nontemporal_load(((const v4f*)pe)  + i);
    v4f b  = __builtin_nontemporal_load(((const v4f*)ps)  + i);
    v4f t  = __builtin_nontemporal_load(((const v4f*)tg)  + i);
    v4i d  = __builtin_nontemporal_load(((const v4i*)ids) + i);
    v4f th = __builtin_nontemporal_load(((const v4f*)thr) + i);
    unsigned cb = __builtin_nontemporal_load(((const unsigned*)cen) + i); // 4 bool bytes

    acc += loss1(a[0], b[0], t[0], d[0],  cb        & 0xFFu, th[0]);
    acc += loss1(a[1], b[1], t[1], d[1], (cb >> 8)  & 0xFFu, th[1]);
    acc += loss1(a[2], b[2], t[2], d[2], (cb >> 16) & 0xFFu, th[2]);
    acc += loss1(a[3], b[3], t[3], d[3], (cb >> 24) & 0xFFu, th[3]);
  }

  // Scalar tail for n % 4 (empty for N = 2^24; reconverges before reduce).
  for (int j = (nv4 << 2) + tid; j < n; j += stride) {
    acc += loss1(pe[j], ps[j], tg[j], ids[j], (unsigned)cen[j], thr[j]);
  }

  const float bsum = block_reduce(acc, lds);
  if (threadIdx.x == 0) partials[blockIdx.x] = bsum;
}

// ---- pass 2: deterministic final reduce + mean ------------------------------
__global__ __launch_bounds__(NTHREADS) void final_reduce(
    const float* __restrict__ partials, float* __restrict__ out, float inv_n) {
  __shared__ float lds[2 * NWAVES];
  float acc = 0.0f;
#pragma unroll
  for (int i = threadIdx.x; i < NBLOCKS; i += NTHREADS) acc += partials[i];
  const float s = block_reduce(acc, lds);
  if (threadIdx.x == 0) out[0] = s * inv_n;
}

extern "C" void kernel_launch(void* const* d_in, const int* in_sizes, int n_in,
                              void* d_out, int out_size, void* d_ws, size_t ws_size,
                              hipStream_t stream) {
  const float*         pe  = (const float*)d_in[0];
  const float*         ps  = (const float*)d_in[1];
  const float*         tg  = (const float*)d_in[2];
  const int*           ids = (const int*)d_in[3];
  const unsigned char* cen = (const unsigned char*)d_in[4];  // JAX bool = 1 byte
  const float*         thr = (const float*)d_in[5];
  float* partials = (float*)d_ws;                            // NBLOCKS floats = 8 KB
  const int n = in_sizes[0];

  dual_head_partial<<<NBLOCKS, NTHREADS, 0, stream>>>(pe, ps, tg, ids, cen, thr,
                                                      partials, n);
  final_reduce<<<1, NTHREADS, 0, stream>>>(partials, (float*)d_out,
                                           1.0f / (float)n);
}